// GCN_37014028157506
// MI455X (gfx1250) — compile-verified
//
#include <hip/hip_runtime.h>

// ---------------------------------------------------------------------------
// GCN forward: softmax( relu(A @ relu(A @ (X@W0)) @ W1) )
// N=100000 nodes, D=256 feat, H1=128, H2=64, E=1.6M edges.
// Memory-bound (~2.7 GB traffic -> ~120us floor at 23.3 TB/s); dense GEMMs
// use fp32 WMMA (V_WMMA_F32_16X16X4_F32) to match reference precision.
// ---------------------------------------------------------------------------

typedef float v2f __attribute__((ext_vector_type(2)));
typedef float v8f __attribute__((ext_vector_type(8)));

#define N_NODES 100000
#define D_FEAT  256
#define H1DIM   128
#define H2DIM   64
#define N_EDGES 1600000

// -------------------------------- zero fill --------------------------------
__global__ __launch_bounds__(256) void zero_kernel(float* __restrict__ p, int n) {
    int i = blockIdx.x * blockDim.x + threadIdx.x;
    if (i < n) p[i] = 0.0f;
}

// ---------------------- GEMM1: H = X @ W0  (fp32 WMMA) ---------------------
// grid = N/16 row tiles; 8 waves/block, wave w owns 16x16 tile at col w*16.
__global__ __launch_bounds__(256) void gemm1_wmma(const float* __restrict__ X,
                                                  const float* __restrict__ W0,
                                                  float* __restrict__ H) {
    const int lane = threadIdx.x & 31;
    const int wave = threadIdx.x >> 5;          // 0..7 -> column tile
    const int row0 = blockIdx.x * 16;
    const int col0 = wave * 16;
    const int m    = lane & 15;                  // M (A) / N (B,C,D)
    const int half = lane >> 4;                  // selects K pair / M+8 half

    v8f acc = {};
    const float* Xrow = X + (size_t)(row0 + m) * D_FEAT;
    for (int k0 = 0; k0 < D_FEAT; k0 += 4) {
        v2f a, b;
        // A 16x4 fp32: VGPR v holds K = v + 2*half  (ISA 7.12.2)
        a.x = Xrow[k0 + 2 * half + 0];
        a.y = Xrow[k0 + 2 * half + 1];
        // B 4x16 fp32: lane%16 = N, VGPR v holds K = v + 2*half
        b.x = W0[(size_t)(k0 + 2 * half + 0) * H1DIM + col0 + m];
        b.y = W0[(size_t)(k0 + 2 * half + 1) * H1DIM + col0 + m];
        acc = __builtin_amdgcn_wmma_f32_16x16x4_f32(false, a, false, b,
                                                    (short)0, acc, false, false);
    }
    // C/D 16x16 f32: VGPR v -> M = v + 8*half, lane%16 -> N
    #pragma unroll
    for (int v = 0; v < 8; ++v) {
        int row = row0 + v + 8 * half;
        H[(size_t)row * H1DIM + col0 + m] = acc[v];
    }
}

// ------------- GEMM2: H2 = relu(A1) @ W1 (ReLU fused into A load) ----------
// grid = N/32; 8 waves/block: wave>>2 picks row tile (2/block), wave&3 col.
__global__ __launch_bounds__(256) void gemm2_wmma(const float* __restrict__ A1,
                                                  const float* __restrict__ W1,
                                                  float* __restrict__ Hout) {
    const int lane = threadIdx.x & 31;
    const int wave = threadIdx.x >> 5;
    const int row0 = (blockIdx.x * 2 + (wave >> 2)) * 16;
    const int col0 = (wave & 3) * 16;
    const int m    = lane & 15;
    const int half = lane >> 4;

    v8f acc = {};
    const float* Arow = A1 + (size_t)(row0 + m) * H1DIM;
    for (int k0 = 0; k0 < H1DIM; k0 += 4) {
        v2f a, b;
        a.x = fmaxf(Arow[k0 + 2 * half + 0], 0.0f);   // fused ReLU(layer-1 agg)
        a.y = fmaxf(Arow[k0 + 2 * half + 1], 0.0f);
        b.x = W1[(size_t)(k0 + 2 * half + 0) * H2DIM + col0 + m];
        b.y = W1[(size_t)(k0 + 2 * half + 1) * H2DIM + col0 + m];
        acc = __builtin_amdgcn_wmma_f32_16x16x4_f32(false, a, false, b,
                                                    (short)0, acc, false, false);
    }
    #pragma unroll
    for (int v = 0; v < 8; ++v) {
        int row = row0 + v + 8 * half;
        Hout[(size_t)row * H2DIM + col0 + m] = acc[v];
    }
}

// ------------------ SpMM: out[dst] += w * Hin[src] (scatter) ---------------
// Each thread handles one edge x one 4-float chunk: 128-bit gather + 4 f32
// atomic adds. A1/A2 (~150MB total) largely resident in 192MB L2.
template <int F>
__global__ __launch_bounds__(256) void spmm_scatter(const int* __restrict__ src,
                                                    const int* __restrict__ dst,
                                                    const float* __restrict__ w,
                                                    const float* __restrict__ Hin,
                                                    float* __restrict__ out) {
    const int lanesPerEdge = F / 4;
    long long tid = (long long)blockIdx.x * blockDim.x + threadIdx.x;
    long long total = (long long)N_EDGES * lanesPerEdge;
    if (tid >= total) return;
    int e = (int)(tid / lanesPerEdge);
    int c = (int)(tid % lanesPerEdge) * 4;
    int s = src[e];
    int d = dst[e];
    float ww = w[e];
    const float4 hv = *(const float4*)(Hin + (size_t)s * F + c);
    float* o = out + (size_t)d * F + c;
    atomicAdd(o + 0, ww * hv.x);
    atomicAdd(o + 1, ww * hv.y);
    atomicAdd(o + 2, ww * hv.z);
    atomicAdd(o + 3, ww * hv.w);
}

// ---------------- ReLU + softmax over 64 classes, wave/node ----------------
__global__ __launch_bounds__(256) void relu_softmax64(const float* __restrict__ A2,
                                                      float* __restrict__ out) {
    const int lane = threadIdx.x & 31;
    const int wave = threadIdx.x >> 5;
    const int node = blockIdx.x * 8 + wave;
    if (node >= N_NODES) return;
    const float* p = A2 + (size_t)node * H2DIM;
    float v0 = fmaxf(p[lane], 0.0f);
    float v1 = fmaxf(p[lane + 32], 0.0f);
    float m = fmaxf(v0, v1);
    #pragma unroll
    for (int off = 16; off >= 1; off >>= 1)
        m = fmaxf(m, __shfl_xor(m, off, 32));
    float e0 = __expf(v0 - m);
    float e1 = __expf(v1 - m);
    float s = e0 + e1;
    #pragma unroll
    for (int off = 16; off >= 1; off >>= 1)
        s += __shfl_xor(s, off, 32);
    float inv = 1.0f / s;
    float* q = out + (size_t)node * H2DIM;
    q[lane]      = e0 * inv;
    q[lane + 32] = e1 * inv;
}

// ---------------------------------------------------------------------------
extern "C" void kernel_launch(void* const* d_in, const int* in_sizes, int n_in,
                              void* d_out, int out_size, void* d_ws, size_t ws_size,
                              hipStream_t stream) {
    (void)in_sizes; (void)n_in; (void)out_size; (void)ws_size;

    const float* x    = (const float*)d_in[0];   // [N, 256]
    const int*   esrc = (const int*)d_in[1];     // [E]
    const int*   edst = (const int*)d_in[2];     // [E]
    const float* ew   = (const float*)d_in[3];   // [E]
    const float* W0   = (const float*)d_in[4];   // [256, 128]
    const float* W1   = (const float*)d_in[5];   // [128, 64]
    float* out = (float*)d_out;                  // [N, 64]

    char* ws = (char*)d_ws;
    float* Hbuf  = (float*)(ws);                                            // N*128
    float* A1    = (float*)(ws + (size_t)N_NODES * H1DIM * 4);              // N*128
    float* H2buf = (float*)(ws + (size_t)N_NODES * H1DIM * 8);              // N*64
    float* A2    = (float*)(ws + (size_t)N_NODES * H1DIM * 8
                               + (size_t)N_NODES * H2DIM * 4);              // N*64

    // Layer 1: H = X @ W0
    gemm1_wmma<<<N_NODES / 16, 256, 0, stream>>>(x, W0, Hbuf);

    // A1 = adj @ H  (zero + edge scatter)
    const int nA1 = N_NODES * H1DIM;
    zero_kernel<<<(nA1 + 255) / 256, 256, 0, stream>>>(A1, nA1);
    const long long t1 = (long long)N_EDGES * (H1DIM / 4);
    spmm_scatter<H1DIM><<<(unsigned)((t1 + 255) / 256), 256, 0, stream>>>(
        esrc, edst, ew, Hbuf, A1);

    // Layer 2: H2 = relu(A1) @ W1   (ReLU fused)
    gemm2_wmma<<<N_NODES / 32, 256, 0, stream>>>(A1, W1, H2buf);

    // A2 = adj @ H2
    const int nA2 = N_NODES * H2DIM;
    zero_kernel<<<(nA2 + 255) / 256, 256, 0, stream>>>(A2, nA2);
    const long long t2 = (long long)N_EDGES * (H2DIM / 4);
    spmm_scatter<H2DIM><<<(unsigned)((t2 + 255) / 256), 256, 0, stream>>>(
        esrc, edst, ew, H2buf, A2);

    // out = softmax(relu(A2), axis=-1)
    relu_softmax64<<<N_NODES / 8, 256, 0, stream>>>(A2, out);
}